// MDLLoss_50629074485397
// MI455X (gfx1250) — compile-verified
//
#include <hip/hip_runtime.h>
#include <math.h>

// ---------------- problem constants (match reference) ----------------
#define N_ROWS 65536
#define N_FEAT 256
#define N_PAR  4194304
#define BLK    1024            // 32 waves (wave32) per feature block
#define EPS5   5.9604645e-07f  // float32 eps * 5
#define H2PE   2.0470955851f   // 0.5*log2(2*pi*e)
#define LAMB   13.2877123795f  // 2*log2(100)
#define INV_LN2 1.4426950408889634f

typedef float v2f __attribute__((ext_vector_type(2)));
typedef float v8f __attribute__((ext_vector_type(8)));

#if defined(__gfx1250__) && __has_builtin(__builtin_amdgcn_wmma_f32_16x16x4_f32)
#define USE_WMMA4 1
#endif

__device__ __forceinline__ bool finitef(float v) {
  // true only for finite values (NaN/Inf fail the compare)
  return __builtin_fabsf(v) <= 3.402823466e+38f;
}

#ifdef USE_WMMA4
// Accumulate 64 values (2 per lane across a wave32) into a 16x16 f32
// accumulator via D = A x ones(4x16) + C.
__device__ __forceinline__ v8f wacc(v8f c, float a0, float a1) {
  v2f a = {a0, a1};
  v2f b = {1.0f, 1.0f};
  return __builtin_amdgcn_wmma_f32_16x16x4_f32(false, a, false, b, (short)0, c,
                                               false, false);
}
// Every column of D carries the same column-sum pattern: lane n holds rows
// 0..7, lane n+16 rows 8..15.  Per-lane 8-term add + xor-16 shuffle gives the
// full wave total in every lane.
__device__ __forceinline__ float wsum(v8f c) {
  float s = c[0] + c[1] + c[2] + c[3] + c[4] + c[5] + c[6] + c[7];
  s += __shfl_xor(s, 16, 32);
  return s;
}
#else
__device__ __forceinline__ float wave_allreduce(float s) {
  for (int m = 16; m > 0; m >>= 1) s += __shfl_xor(s, m, 32);
  return s;
}
#endif

// ---------------- two-parameter Yeo-Johnson ----------------
__device__ __forceinline__ void yj_eval(float x, float l1, float l2,
                                        float& y, float& lj) {
  if (x >= 0.0f) {
    float lp = log1pf(x);
    if (__builtin_fabsf(l1) < EPS5) { y = lp; lj = -lp; }
    else { y = expm1f(l1 * lp) / l1; lj = (l1 - 1.0f) * lp; }
  } else {
    if (__builtin_fabsf(l2 - 2.0f) < EPS5) {
      float ln = log1pf(-x); y = -ln; lj = -ln;
    } else {
      float tm = 2.0f - l2;
      float base = fmaxf(1.0f - x, 1e-18f);
      float lb = logf(base);
      y = -expm1f(tm * lb) / tm;
      lj = (1.0f - l2) * lb;
    }
  }
}

__device__ __forceinline__ float feature_bits_f(float nn, float nf, float sy,
                                                float sy2, float slj,
                                                float res) {
  float nex = nn - nf;
  float mean = sy / fmaxf(nf, 1.0f);
  float var = fmaxf((sy2 - nf * mean * mean) / fmaxf(nf, 1.0f), 1e-12f);
  float db = (nf > 1.0f) ? nf * (H2PE + 0.5f * log2f(var)) : 0.0f;
  float ljb = (nf > 0.0f) ? slj * INV_LN2 : 0.0f;
  float exb = (nex > 0.0f) ? nex * log2f(1.0f / res) : 0.0f;
  float pb = 0.0f;
  if (nex > 0.0f && nex < nn) {
    pb = (lgammaf(nn + 1.0f) - lgammaf(nex + 1.0f) - lgammaf(nn - nex + 1.0f)) *
         INV_LN2;
  }
  return db + ljb + exb + pb + LAMB;
}

// ---------------- in-LDS bitonic sort (ascending, 64K elems) ----------------
__device__ void bitonic_sort(float* col, int tid) {
  for (unsigned k = 2; k <= N_ROWS; k <<= 1) {
    for (unsigned j = k >> 1; j > 0; j >>= 1) {
      for (unsigned i = tid; i < N_ROWS; i += BLK) {
        unsigned ixj = i ^ j;
        if (ixj > i) {
          float a = col[i], b = col[ixj];
          bool up = ((i & k) == 0);
          bool sw = up ? (a > b) : (a < b);
          if (sw) { col[i] = b; col[ixj] = a; }
        }
      }
      __syncthreads();
    }
  }
}

// one masked-reduction pass over the LDS-resident column; leaves
// {cnt, Σy, Σy², Σlj} in red[128..131] (thread 0's view after the barrier)
__device__ void pass_sums(const float* col, float l1, float l2, float* red,
                          int tid) {
#ifdef USE_WMMA4
  v8f z = {0.f, 0.f, 0.f, 0.f, 0.f, 0.f, 0.f, 0.f};
  v8f act = z, asy = z, asy2 = z, aslj = z;
#else
  float act = 0.f, asy = 0.f, asy2 = 0.f, aslj = 0.f;
#endif
  for (int e = 0; e < 64; e += 2) {
    float v0 = col[e * BLK + tid];
    float v1 = col[(e + 1) * BLK + tid];
    float y0, lj0, y1, lj1;
    yj_eval(v0, l1, l2, y0, lj0);
    yj_eval(v1, l1, l2, y1, lj1);
    bool m0 = finitef(v0) && finitef(y0);
    bool m1 = finitef(v1) && finitef(y1);
    float c0 = m0 ? 1.f : 0.f, c1 = m1 ? 1.f : 0.f;
    y0 = m0 ? y0 : 0.f; lj0 = m0 ? lj0 : 0.f;
    y1 = m1 ? y1 : 0.f; lj1 = m1 ? lj1 : 0.f;
#ifdef USE_WMMA4
    act  = wacc(act, c0, c1);
    asy  = wacc(asy, y0, y1);
    asy2 = wacc(asy2, y0 * y0, y1 * y1);
    aslj = wacc(aslj, lj0, lj1);
#else
    act += c0 + c1; asy += y0 + y1;
    asy2 += y0 * y0 + y1 * y1; aslj += lj0 + lj1;
#endif
  }
#ifdef USE_WMMA4
  float tct = wsum(act), tsy = wsum(asy), tsy2 = wsum(asy2), tslj = wsum(aslj);
#else
  float tct = wave_allreduce(act), tsy = wave_allreduce(asy);
  float tsy2 = wave_allreduce(asy2), tslj = wave_allreduce(aslj);
#endif
  int lane = tid & 31, w = tid >> 5;
  if (lane == 0) {
    red[w] = tct; red[32 + w] = tsy; red[64 + w] = tsy2; red[96 + w] = tslj;
  }
  __syncthreads();
  if (tid == 0) {
    float c = 0.f, s = 0.f, s2 = 0.f, l = 0.f;
    for (int i = 0; i < 32; ++i) {
      c += red[i]; s += red[32 + i]; s2 += red[64 + i]; l += red[96 + i];
    }
    red[128] = c; red[129] = s; red[130] = s2; red[131] = l;
  }
  __syncthreads();
}

// ---------------- per-feature pipeline: one block per feature ----------------
__global__ __launch_bounds__(BLK) void mdl_feature_kernel(
    const float* __restrict__ x, const float* __restrict__ lam1,
    const float* __restrict__ lam2, const float* __restrict__ rp,
    float* __restrict__ feat_bits) {
  extern __shared__ float smem[];
  float* col = smem;            // 65536 floats (256 KB)
  float* red = smem + N_ROWS;   // 160 floats
  float* scr = red + 160;       // 16 floats: scores[6], res, best l1/l2
  __shared__ int s_cnt[2];
  const int f = blockIdx.x;
  const int tid = threadIdx.x;
  if (tid == 0) { s_cnt[0] = 0; s_cnt[1] = 0; }
  __syncthreads();

  // phase 1: load column, nonfinite -> +inf for sorting, count finite
  int mx = 0;
  for (int e = 0; e < 64; ++e) {
    int i = e * BLK + tid;
    float v = x[(size_t)i * N_FEAT + f];
    bool fn = finitef(v);
    col[i] = fn ? v : __builtin_inff();
    mx += fn;
  }
  atomicAdd(&s_cnt[0], mx);
  __syncthreads();

  // phase 2: sort column
  bitonic_sort(col, tid);

  // phase 3: gaps -> qualifying diffs (else +inf), count them
  float qv[64];
  int mq = 0;
#pragma unroll
  for (int e = 0; e < 64; ++e) {
    int i = e * BLK + tid;
    float d = (i < N_ROWS - 1) ? (col[i + 1] - col[i]) : __builtin_inff();
    bool g = finitef(d) && (d > 1e-12f);
    qv[e] = g ? d : __builtin_inff();
    mq += g;
  }
  __syncthreads();
#pragma unroll
  for (int e = 0; e < 64; ++e) col[e * BLK + tid] = qv[e];
  atomicAdd(&s_cnt[1], mq);
  __syncthreads();

  // phase 4: sort diffs, take lower median -> resolution
  bitonic_sort(col, tid);
  if (tid == 0) {
    float r;
    if (s_cnt[0] < 2 || s_cnt[1] == 0) r = 1e-6f;
    else {
      int idx = (s_cnt[1] - 1) >> 1;
      r = fminf(fmaxf(col[idx], 1e-12f), 0.1f);
    }
    scr[6] = r;
  }
  __syncthreads();

  // phase 5: reload raw column into LDS (order no longer matters for sums)
  for (int e = 0; e < 64; ++e) {
    int i = e * BLK + tid;
    col[i] = x[(size_t)i * N_FEAT + f];
  }
  __syncthreads();

  float L1[6], L2[6];
  L1[0] = lam1[f]; L2[0] = lam2[f];
#pragma unroll
  for (int c = 1; c < 6; ++c) {
    L1[c] = L1[0] + 0.1f * rp[f * 10 + 2 * (c - 1)];
    L2[c] = L2[0] + 0.1f * rp[f * 10 + 2 * (c - 1) + 1];
  }
  float res = scr[6];

  // phase 6: six candidate scores
  for (int c = 0; c < 6; ++c) {
    pass_sums(col, L1[c], L2[c], red, tid);
    if (tid == 0)
      scr[c] = feature_bits_f((float)N_ROWS, red[128], red[129], red[130],
                              red[131], res);
    __syncthreads();
  }

  // phase 7: tie-averaged argmin, final scoring pass
  if (tid == 0) {
    float mn = scr[0];
    for (int c = 1; c < 6; ++c) mn = fminf(mn, scr[c]);
    float tw = 0.f, b1 = 0.f, b2 = 0.f;
    for (int c = 0; c < 6; ++c)
      if (scr[c] == mn) { tw += 1.f; b1 += L1[c]; b2 += L2[c]; }
    tw = fmaxf(tw, 1e-8f);
    scr[8] = b1 / tw; scr[9] = b2 / tw;
  }
  __syncthreads();

  pass_sums(col, scr[8], scr[9], red, tid);
  if (tid == 0)
    feat_bits[f] = feature_bits_f((float)N_ROWS, red[128], red[129], red[130],
                                  red[131], res);
}

// ---------------- params tensor partial reduction (WMMA stream) -------------
__global__ __launch_bounds__(256) void mdl_model_partial(
    const float* __restrict__ p, float* __restrict__ out3) {
  __shared__ float red[24];
  const int g = blockIdx.x * 256 + threadIdx.x;  // 512*256 = 131072 threads
#ifdef USE_WMMA4
  v8f z = {0.f, 0.f, 0.f, 0.f, 0.f, 0.f, 0.f, 0.f};
  v8f act = z, asy = z, asy2 = z;
#else
  float act = 0.f, asy = 0.f, asy2 = 0.f;
#endif
  for (int e = 0; e < 32; e += 2) {  // 32 * 131072 == N_PAR exactly
    float v0 = p[(size_t)e * 131072 + g];
    float v1 = p[(size_t)(e + 1) * 131072 + g];
    bool m0 = finitef(v0), m1 = finitef(v1);
    float a0 = m0 ? v0 : 0.f, a1 = m1 ? v1 : 0.f;
#ifdef USE_WMMA4
    act  = wacc(act, m0 ? 1.f : 0.f, m1 ? 1.f : 0.f);
    asy  = wacc(asy, a0, a1);
    asy2 = wacc(asy2, a0 * a0, a1 * a1);
#else
    act += (m0 ? 1.f : 0.f) + (m1 ? 1.f : 0.f);
    asy += a0 + a1; asy2 += a0 * a0 + a1 * a1;
#endif
  }
#ifdef USE_WMMA4
  float tct = wsum(act), tsy = wsum(asy), tsy2 = wsum(asy2);
#else
  float tct = wave_allreduce(act), tsy = wave_allreduce(asy);
  float tsy2 = wave_allreduce(asy2);
#endif
  int lane = threadIdx.x & 31, w = threadIdx.x >> 5;
  if (lane == 0) { red[w] = tct; red[8 + w] = tsy; red[16 + w] = tsy2; }
  __syncthreads();
  if (threadIdx.x == 0) {
    float c = 0.f, s = 0.f, s2 = 0.f;
    for (int i = 0; i < 8; ++i) { c += red[i]; s += red[8 + i]; s2 += red[16 + i]; }
    out3[blockIdx.x * 3 + 0] = c;
    out3[blockIdx.x * 3 + 1] = s;
    out3[blockIdx.x * 3 + 2] = s2;
  }
}

// ---------------- deterministic final fold ----------------
__global__ void mdl_final(const float* __restrict__ fb,
                          const float* __restrict__ mp,
                          float* __restrict__ out) {
  if (threadIdx.x == 0 && blockIdx.x == 0) {
    float data = 0.f;
    for (int i = 0; i < N_FEAT; ++i) data += fb[i];
    float c = 0.f, s = 0.f, s2 = 0.f;
    for (int b = 0; b < 512; ++b) {
      c += mp[3 * b]; s += mp[3 * b + 1]; s2 += mp[3 * b + 2];
    }
    float model = feature_bits_f((float)N_PAR, c, s, s2, 0.0f, 1e-12f);
    out[0] = data + model;
  }
}

extern "C" void kernel_launch(void* const* d_in, const int* in_sizes, int n_in,
                              void* d_out, int out_size, void* d_ws,
                              size_t ws_size, hipStream_t stream) {
  (void)in_sizes; (void)n_in; (void)out_size; (void)ws_size;
  const float* x      = (const float*)d_in[0];
  const float* lam1   = (const float*)d_in[1];
  const float* lam2   = (const float*)d_in[2];
  const float* rp     = (const float*)d_in[3];
  const float* params = (const float*)d_in[4];
  float* feat = (float*)d_ws;       // 256 floats: per-feature bits
  float* mp   = feat + N_FEAT;      // 512*3 floats: model partials

  size_t smem = (size_t)(N_ROWS + 160 + 16) * sizeof(float);  // ~256.7 KB LDS
  mdl_feature_kernel<<<N_FEAT, BLK, smem, stream>>>(x, lam1, lam2, rp, feat);
  mdl_model_partial<<<512, 256, 0, stream>>>(params, mp);
  mdl_final<<<1, 32, 0, stream>>>(feat, mp, (float*)d_out);
}